// GAT_71708773974792
// MI455X (gfx1250) — compile-verified
//
#include <hip/hip_runtime.h>
#include <math.h>

// ---------------- problem constants (from reference) ----------------
#define NNODES 50000
#define NEDGES 800000
#define ETOT   (NNODES + NEDGES)   // 850000 (self-loops appended)
#define INC    128
#define DIMC   64
#define H1     8
#define HC1    (H1 * DIMC)         // 512
#define NGR    512
#define OUTC   10
#define OUTP   32                  // padded head width (multiple of 32)
#define NEG_SLOPE 0.2f

typedef float v2f __attribute__((ext_vector_type(2)));
typedef float v8f __attribute__((ext_vector_type(8)));

// =====================================================================
// fp32 WMMA GEMM:  C[M x Nc] = A[M x K] @ B[K x Nc], row-major.
// Requirements: M % 16 == 0, Nc % 32 == 0, K % 4 == 0 (all hold here;
// the ragged 10-wide head GEMM runs on a zero-padded 32-wide B/C).
// One wave per 16x32 C tile: two v8f accumulators, 2 WMMAs per k-step,
// one shared A fragment (global_load_b64) amortized over both.
// No guards anywhere -> EXEC all-ones at every WMMA, no exec churn.
// =====================================================================
__global__ void gemm_wmma_f32(const float* __restrict__ A,
                              const float* __restrict__ B,
                              float* __restrict__ C,
                              int M, int K, int Nc) {
  const int tilesN = Nc >> 5;
  const int tilesM = M >> 4;
  const int wave = (blockIdx.x * blockDim.x + threadIdx.x) >> 5;
  if (wave >= tilesM * tilesN) return;          // wave-uniform
  const int lane = threadIdx.x & 31;
  const int tm = wave / tilesN;
  const int tn = wave % tilesN;
  const int r  = lane & 15;                     // m (for A) / n (for B)
  const int kh = (lane >> 4) << 1;              // 0 or 2: K sub-offset

  v8f acc0 = {0.f,0.f,0.f,0.f,0.f,0.f,0.f,0.f};
  v8f acc1 = {0.f,0.f,0.f,0.f,0.f,0.f,0.f,0.f};

  const float* ap  = A + (size_t)(tm * 16 + r) * K + kh;    // 8B aligned
  const float* b00 = B + (size_t)kh * Nc + tn * 32 + r;     // (kh  , n0)
  const float* b01 = b00 + Nc;                              // (kh+1, n0)
  const float* b10 = b00 + 16;                              // (kh  , n1)
  const float* b11 = b01 + 16;                              // (kh+1, n1)
  const size_t bstep = (size_t)Nc * 4;

  for (int k0 = 0; k0 < K; k0 += 4) {
    v2f a = *reinterpret_cast<const v2f*>(ap);
    v2f b0, b1;
    b0.x = *b00; b0.y = *b01;
    b1.x = *b10; b1.y = *b11;
    acc0 = __builtin_amdgcn_wmma_f32_16x16x4_f32(false, a, false, b0,
                                                 (short)0, acc0, false, false);
    acc1 = __builtin_amdgcn_wmma_f32_16x16x4_f32(false, a, false, b1,
                                                 (short)0, acc1, false, false);
    ap += 4; b00 += bstep; b01 += bstep; b10 += bstep; b11 += bstep;
  }

  // C/D layout: vgpr i -> row (lane>=16 ? 8 : 0)+i, col lane&15
  const int cm = tm * 16 + ((lane >> 4) << 3);
  float* cp = C + (size_t)cm * Nc + tn * 32 + r;
  #pragma unroll
  for (int i = 0; i < 8; ++i) {
    cp[(size_t)i * Nc]      = acc0[i];
    cp[(size_t)i * Nc + 16] = acc1[i];
  }
}

// ---------------- elementwise helpers ----------------
__global__ void fill_f32(float* __restrict__ p, float v, int n) {
  int i = blockIdx.x * blockDim.x + threadIdx.x;
  if (i < n) p[i] = v;
}

__global__ void bias_elu_kernel(float* __restrict__ x, const float* __restrict__ b,
                                int total, int cols) {
  int i = blockIdx.x * blockDim.x + threadIdx.x;
  if (i >= total) return;
  float v = x[i] + b[i % cols];
  x[i] = v > 0.f ? v : expm1f(v);
}

__global__ void bias_relu_kernel(float* __restrict__ x, const float* __restrict__ b,
                                 int total, int cols) {
  int i = blockIdx.x * blockDim.x + threadIdx.x;
  if (i >= total) return;
  float v = x[i] + b[i % cols];
  x[i] = v > 0.f ? v : 0.f;
}

__global__ void bias_add_kernel(float* __restrict__ x, const float* __restrict__ b,
                                int total, int cols) {
  int i = blockIdx.x * blockDim.x + threadIdx.x;
  if (i >= total) return;
  x[i] += b[i % cols];
}

// pad lw2 [64,10] -> [64,32] with zeros so the head GEMM is unguarded
__global__ void pad_lw2_kernel(const float* __restrict__ src, float* __restrict__ dst) {
  int i = blockIdx.x * blockDim.x + threadIdx.x;   // 64*32
  if (i >= DIMC * OUTP) return;
  int k = i >> 5, n = i & (OUTP - 1);
  dst[i] = (n < OUTC) ? src[k * OUTC + n] : 0.f;
}

// out[512,10] = outp[512,32][:, :10] + lb2
__global__ void final_out_kernel(const float* __restrict__ outp,
                                 const float* __restrict__ lb2,
                                 float* __restrict__ out) {
  int i = blockIdx.x * blockDim.x + threadIdx.x;   // 512*10
  if (i >= NGR * OUTC) return;
  int row = i / OUTC, c = i - row * OUTC;
  out[i] = outp[row * OUTP + c] + lb2[c];
}

// ---------------- attention pieces ----------------
template <int H>
__global__ void alpha_kernel(const float* __restrict__ h,
                             const float* __restrict__ a_src,
                             const float* __restrict__ a_dst,
                             float* __restrict__ as, float* __restrict__ ad) {
  int idx = blockIdx.x * blockDim.x + threadIdx.x;   // idx = n*H + hh
  if (idx >= NNODES * H) return;
  int hh = idx % H;
  const float* hp = h + (size_t)idx * DIMC;
  float s = 0.f, d = 0.f;
  #pragma unroll 8
  for (int c = 0; c < DIMC; ++c) {
    float v = hp[c];
    s += v * a_src[hh * DIMC + c];
    d += v * a_dst[hh * DIMC + c];
  }
  as[idx] = s;
  ad[idx] = d;
}

__device__ __forceinline__ void edge_endpoints(int e, const int* __restrict__ ei,
                                               int& s, int& d) {
  if (e < NEDGES) { s = ei[e]; d = ei[NEDGES + e]; }
  else            { s = e - NEDGES; d = s; }          // self-loops
}

__device__ __forceinline__ float lrelu(float v) {
  return v > 0.f ? v : NEG_SLOPE * v;
}

template <int H>
__global__ void edge_max_kernel(const int* __restrict__ ei,
                                const float* __restrict__ as,
                                const float* __restrict__ ad,
                                float* __restrict__ emax) {
  int t = blockIdx.x * blockDim.x + threadIdx.x;
  if (t >= ETOT * H) return;
  int e = t / H, hh = t - e * H;
  int s, d; edge_endpoints(e, ei, s, d);
  float v = lrelu(as[s * H + hh] + ad[d * H + hh]);
  atomicMax(&emax[d * H + hh], v);       // global_atomic_max_num_f32
}

template <int H>
__global__ void edge_sum_kernel(const int* __restrict__ ei,
                                const float* __restrict__ as,
                                const float* __restrict__ ad,
                                const float* __restrict__ emax,
                                float* __restrict__ den) {
  int t = blockIdx.x * blockDim.x + threadIdx.x;
  if (t >= ETOT * H) return;
  int e = t / H, hh = t - e * H;
  int s, d; edge_endpoints(e, ei, s, d);
  float v = lrelu(as[s * H + hh] + ad[d * H + hh]);
  atomicAdd(&den[d * H + hh], expf(v - emax[d * H + hh]));
}

// One wave32 per edge. Lanes 0..H-1 compute the softmax weight for their
// head once; __shfl broadcasts it to the lanes streaming that head's
// channels. Gather of h[src] and atomic scatter to out[dst] are fully
// coalesced (32-lane stride-32 sweeps over H*64 channels).
template <int H>
__global__ void agg_kernel(const int* __restrict__ ei,
                           const float* __restrict__ as,
                           const float* __restrict__ ad,
                           const float* __restrict__ emax,
                           const float* __restrict__ den,
                           const float* __restrict__ h,
                           float* __restrict__ out) {
  int wave = (blockIdx.x * blockDim.x + threadIdx.x) >> 5;
  if (wave >= ETOT) return;
  int lane = threadIdx.x & 31;
  int s, d; edge_endpoints(wave, ei, s, d);

  float myc = 0.f;
  if (lane < H) {
    float v = lrelu(as[s * H + lane] + ad[d * H + lane]);
    myc = expf(v - emax[d * H + lane]) / den[d * H + lane];
  }

  const float* hs = h + (size_t)s * (H * DIMC);
  float* od = out + (size_t)d * (H * DIMC);
  #pragma unroll
  for (int c = lane; c < H * DIMC; c += 32) {
    float coef = __shfl(myc, c >> 6, 32);
    atomicAdd(&od[c], coef * hs[c]);
  }
}

// global_add_pool: g[batch[n], c] += h[n, c]
__global__ void pool_kernel(const float* __restrict__ h,
                            const int* __restrict__ batch,
                            float* __restrict__ g) {
  int i = blockIdx.x * blockDim.x + threadIdx.x;
  if (i >= NNODES * DIMC) return;
  int n = i >> 6, c = i & 63;
  atomicAdd(&g[batch[n] * DIMC + c], h[i]);
}

// =====================================================================
extern "C" void kernel_launch(void* const* d_in, const int* in_sizes, int n_in,
                              void* d_out, int out_size, void* d_ws, size_t ws_size,
                              hipStream_t stream) {
  const float* x      = (const float*)d_in[0];
  const int*   ei     = (const int*)  d_in[1];
  const int*   batch  = (const int*)  d_in[2];
  const float* W1     = (const float*)d_in[3];
  const float* a_src1 = (const float*)d_in[4];
  const float* a_dst1 = (const float*)d_in[5];
  const float* b1     = (const float*)d_in[6];
  const float* W2     = (const float*)d_in[7];
  const float* a_src2 = (const float*)d_in[8];
  const float* a_dst2 = (const float*)d_in[9];
  const float* b2     = (const float*)d_in[10];
  const float* lw1    = (const float*)d_in[11];
  const float* lb1    = (const float*)d_in[12];
  const float* lw2    = (const float*)d_in[13];
  const float* lb2    = (const float*)d_in[14];
  float* out = (float*)d_out;

  // ---- workspace carve-out (~235 MB) ----
  char* ws = (char*)d_ws;
  size_t off = 0;
  auto alloc = [&](size_t nfloats) {
    float* p = (float*)(ws + off);
    off += ((nfloats * sizeof(float) + 255) / 256) * 256;
    return p;
  };
  float* h1   = alloc((size_t)NNODES * HC1);   // x @ W1          [N,512]
  float* o1   = alloc((size_t)NNODES * HC1);   // conv1 out       [N,512]
  float* as1  = alloc((size_t)NNODES * H1);
  float* ad1  = alloc((size_t)NNODES * H1);
  float* em1  = alloc((size_t)NNODES * H1);
  float* dn1  = alloc((size_t)NNODES * H1);
  float* h2   = alloc((size_t)NNODES * DIMC);  // o1 @ W2         [N,64]
  float* o2   = alloc((size_t)NNODES * DIMC);  // conv2 out       [N,64]
  float* as2  = alloc(NNODES);
  float* ad2  = alloc(NNODES);
  float* em2  = alloc(NNODES);
  float* dn2  = alloc(NNODES);
  float* g    = alloc((size_t)NGR * DIMC);     // pooled          [512,64]
  float* gr   = alloc((size_t)NGR * DIMC);     // mlp hidden      [512,64]
  float* lw2p = alloc((size_t)DIMC * OUTP);    // padded lw2      [64,32]
  float* outp = alloc((size_t)NGR * OUTP);     // padded head out [512,32]

  const int T = 256;
  auto blk1 = [](int n) { return (n + 255) / 256; };
  auto gemm = [&](const float* A, const float* B, float* C, int M, int K, int N) {
    int waves = (M >> 4) * (N >> 5);
    gemm_wmma_f32<<<(waves + 7) / 8, T, 0, stream>>>(A, B, C, M, K, N);
  };

  // ---- init accumulators ----
  hipMemsetAsync(o1,  0, (size_t)NNODES * HC1  * 4, stream);
  hipMemsetAsync(dn1, 0, (size_t)NNODES * H1   * 4, stream);
  hipMemsetAsync(o2,  0, (size_t)NNODES * DIMC * 4, stream);
  hipMemsetAsync(dn2, 0, (size_t)NNODES        * 4, stream);
  hipMemsetAsync(g,   0, (size_t)NGR   * DIMC  * 4, stream);
  fill_f32<<<blk1(NNODES * H1), T, 0, stream>>>(em1, -INFINITY, NNODES * H1);
  fill_f32<<<blk1(NNODES),      T, 0, stream>>>(em2, -INFINITY, NNODES);
  pad_lw2_kernel<<<blk1(DIMC * OUTP), T, 0, stream>>>(lw2, lw2p);

  // ---- conv1: GATConv(128 -> 64, heads=8, concat) ----
  gemm(x, W1, h1, NNODES, INC, HC1);                       // h1 = x @ W1
  alpha_kernel<H1><<<blk1(NNODES * H1), T, 0, stream>>>(h1, a_src1, a_dst1, as1, ad1);
  edge_max_kernel<H1><<<blk1(ETOT * H1), T, 0, stream>>>(ei, as1, ad1, em1);
  edge_sum_kernel<H1><<<blk1(ETOT * H1), T, 0, stream>>>(ei, as1, ad1, em1, dn1);
  agg_kernel<H1><<<(ETOT * 32 + T - 1) / T, T, 0, stream>>>(ei, as1, ad1, em1, dn1, h1, o1);
  bias_elu_kernel<<<blk1(NNODES * HC1), T, 0, stream>>>(o1, b1, NNODES * HC1, HC1);

  // ---- conv2: GATConv(512 -> 64, heads=1, concat=False) ----
  gemm(o1, W2, h2, NNODES, HC1, DIMC);                     // h2 = elu(o1) @ W2
  alpha_kernel<1><<<blk1(NNODES), T, 0, stream>>>(h2, a_src2, a_dst2, as2, ad2);
  edge_max_kernel<1><<<blk1(ETOT), T, 0, stream>>>(ei, as2, ad2, em2);
  edge_sum_kernel<1><<<blk1(ETOT), T, 0, stream>>>(ei, as2, ad2, em2, dn2);
  agg_kernel<1><<<(ETOT * 32 + T - 1) / T, T, 0, stream>>>(ei, as2, ad2, em2, dn2, h2, o2);
  bias_add_kernel<<<blk1(NNODES * DIMC), T, 0, stream>>>(o2, b2, NNODES * DIMC, DIMC);

  // ---- global_add_pool + MLP head ----
  pool_kernel<<<blk1(NNODES * DIMC), T, 0, stream>>>(o2, batch, g);
  gemm(g, lw1, gr, NGR, DIMC, DIMC);                       // gr = g @ lw1
  bias_relu_kernel<<<blk1(NGR * DIMC), T, 0, stream>>>(gr, lb1, NGR * DIMC, DIMC);
  gemm(gr, lw2p, outp, NGR, DIMC, OUTP);                   // outp = gr @ lw2p
  final_out_kernel<<<blk1(NGR * OUTC), T, 0, stream>>>(outp, lb2, out);
}